// LinearMultiheadAttention_75204877353238
// MI455X (gfx1250) — compile-verified
//
#include <hip/hip_runtime.h>

// B=4, S=8192, D=256, H=8 ; f32 in / f32 out
#define B_ 4
#define S_ 8192
#define D_ 256
#define H_ 8
#define LOG2E 1.4426950408889634f

// workspace layout (float indices)
#define OFF_K     0      // 8   : Kh[h]   = log2(e) * sum(q_weights[h])
#define OFF_AC    8      // 8   : Ac[h]   = D * sum(v_weights[h])
#define OFF_VSUM  16     // 1024: Vsum[b][d]
#define OFF_QMAXE 1040   // 1024 (uint, order-encoded)
#define OFF_QMINE 2064   // 1024 (uint, order-encoded)
#define OFF_M2    3088   // 8192: m2[b][h][d] = max_s(Kh*q)
#define OFF_Z     11280  // 8192: Z[b][h][d] = sum_s exp2(Kh*q - m2)
#define OFF_T     19472  // 8192: t = m2 + log2(Z)

typedef float v2f __attribute__((ext_vector_type(2)));
typedef float v8f __attribute__((ext_vector_type(8)));

// monotonic float<->uint order encoding (for deterministic atomic max/min)
__device__ __forceinline__ unsigned f2ord(float f) {
  unsigned u = __float_as_uint(f);
  return (u & 0x80000000u) ? ~u : (u | 0x80000000u);
}
__device__ __forceinline__ float ord2f(unsigned e) {
  unsigned u = (e & 0x80000000u) ? (e & 0x7FFFFFFFu) : ~e;
  return __uint_as_float(u);
}

__device__ __forceinline__ float fast_exp2(float x) {
#if __has_builtin(__builtin_amdgcn_exp2f)
  return __builtin_amdgcn_exp2f(x);   // v_exp_f32
#else
  return exp2f(x);
#endif
}
__device__ __forceinline__ float fast_log2(float x) {
#if __has_builtin(__builtin_amdgcn_logf)
  return __builtin_amdgcn_logf(x);    // v_log_f32
#else
  return log2f(x);
#endif
}

// ---- K0: init reduction buffers ---------------------------------------------
__global__ void k0_init(float* __restrict__ ws) {
  int i = blockIdx.x * blockDim.x + threadIdx.x;
  if (i >= OFF_T) return;
  if (i >= OFF_QMAXE && i < OFF_QMINE)      ((unsigned*)ws)[i] = 0x007FFFFFu; // enc(-inf)
  else if (i >= OFF_QMINE && i < OFF_M2)    ((unsigned*)ws)[i] = 0xFF800000u; // enc(+inf)
  else                                      ws[i] = 0.0f;
}

// ---- K1: weight sums via WMMA ones-trick ------------------------------------
// Each 16x16x4 f32 WMMA with B==1 accumulates 64 streamed elements into row
// sums; every accumulator column then holds the same row-sum, so
// total = (sum of 8 D-components) + shfl_xor(.,16).
__global__ void k1_wsum(const float* __restrict__ qw, const float* __restrict__ vw,
                        float* __restrict__ ws) {
  const int h   = blockIdx.x & 7;
  const int isV = blockIdx.x >> 3;
  const float* src = (isV ? vw : qw) + (size_t)h * (D_ * D_);
  const int wave = threadIdx.x >> 5, lane = threadIdx.x & 31;
  float s;
#if defined(__gfx1250__) && __has_builtin(__builtin_amdgcn_wmma_f32_16x16x4_f32)
  v8f acc = {};
  const v2f ones = {1.0f, 1.0f};
  for (int it = 0; it < 128; ++it) {
    int off = ((it * 8 + wave) << 6) + (lane << 1);
    v2f a = *(const v2f*)(src + off);
    acc = __builtin_amdgcn_wmma_f32_16x16x4_f32(false, a, false, ones,
                                                (short)0, acc, false, false);
  }
  s = acc[0] + acc[1] + acc[2] + acc[3] + acc[4] + acc[5] + acc[6] + acc[7];
  s += __shfl_xor(s, 16, 32);
#else
  float p = 0.0f;
  for (int it = 0; it < 128; ++it) {
    int off = ((it * 8 + wave) << 6) + (lane << 1);
    v2f a = *(const v2f*)(src + off);
    p += a[0] + a[1];
  }
  s = p;
  for (int o = 16; o; o >>= 1) s += __shfl_xor(s, o, 32);
#endif
  __shared__ float red[8];
  if (lane == 0) red[wave] = s;
  __syncthreads();
  if (threadIdx.x == 0) {
    float tot = 0.0f;
    for (int w = 0; w < 8; ++w) tot += red[w];
    if (isV) ws[OFF_AC + h] = (float)D_ * tot;
    else     ws[OFF_K + h]  = LOG2E * tot;
  }
}

// ---- K2: per-(b,d) q max/min + Vsum (streams q and v once) ------------------
#define K2_CHUNKS 64
#define K2_ROWS   (S_ / K2_CHUNKS)
__global__ void k2_stats(const float* __restrict__ q, const float* __restrict__ v,
                         float* __restrict__ ws) {
  const int b  = blockIdx.x / K2_CHUNKS;
  const int ch = blockIdx.x % K2_CHUNKS;
  const int d  = threadIdx.x;
  const float* qp = q + ((size_t)(b * S_ + ch * K2_ROWS)) * D_ + d;
  const float* vp = v + ((size_t)(b * S_ + ch * K2_ROWS)) * D_ + d;
  float qmx = -3.402823466e38f, qmn = 3.402823466e38f, vs = 0.0f;
  for (int r = 0; r < K2_ROWS; ++r) {
    if (r + 8 < K2_ROWS) {
      __builtin_prefetch(qp + (size_t)(r + 8) * D_, 0, 1);
      __builtin_prefetch(vp + (size_t)(r + 8) * D_, 0, 1);
    }
    float qv = qp[(size_t)r * D_];
    float vv = vp[(size_t)r * D_];
    qmx = fmaxf(qmx, qv);
    qmn = fminf(qmn, qv);
    vs += vv;
  }
  unsigned* qmaxE = (unsigned*)(ws + OFF_QMAXE);
  unsigned* qminE = (unsigned*)(ws + OFF_QMINE);
  atomicMax(&qmaxE[b * D_ + d], f2ord(qmx));
  atomicMin(&qminE[b * D_ + d], f2ord(qmn));
  atomicAdd(&ws[OFF_VSUM + b * D_ + d], vs);
}

// ---- K3: m2[b,h,d] = max_s(Kh * q) ------------------------------------------
__global__ void k3_prep(float* __restrict__ ws) {
  int i = blockIdx.x * blockDim.x + threadIdx.x;
  if (i >= B_ * H_ * D_) return;
  int d = i & (D_ - 1);
  int h = (i >> 8) & (H_ - 1);
  int b = i >> 11;
  float K = ws[OFF_K + h];
  unsigned* qmaxE = (unsigned*)(ws + OFF_QMAXE);
  unsigned* qminE = (unsigned*)(ws + OFF_QMINE);
  float qmx = ord2f(qmaxE[b * D_ + d]);
  float qmn = ord2f(qminE[b * D_ + d]);
  ws[OFF_M2 + i] = (K >= 0.0f) ? K * qmx : K * qmn;
}

// ---- K4: Z[b,h,d] = sum_s exp2(Kh*q - m2) (streams q) -----------------------
#define K4_CHUNKS 32
#define K4_ROWS   (S_ / K4_CHUNKS)
__global__ void k4_z(const float* __restrict__ q, float* __restrict__ ws) {
  const int b  = blockIdx.x / K4_CHUNKS;
  const int ch = blockIdx.x % K4_CHUNKS;
  const int d  = threadIdx.x;
  float K[H_], m2[H_], z[H_];
#pragma unroll
  for (int h = 0; h < H_; ++h) {
    K[h]  = ws[OFF_K + h];
    m2[h] = ws[OFF_M2 + ((b * H_ + h) << 8) + d];
    z[h]  = 0.0f;
  }
  const float* qp = q + ((size_t)(b * S_ + ch * K4_ROWS)) * D_ + d;
  for (int r = 0; r < K4_ROWS; ++r) {
    if (r + 8 < K4_ROWS) __builtin_prefetch(qp + (size_t)(r + 8) * D_, 0, 1);
    float qv = qp[(size_t)r * D_];
#pragma unroll
    for (int h = 0; h < H_; ++h) z[h] += fast_exp2(fmaf(K[h], qv, -m2[h]));
  }
#pragma unroll
  for (int h = 0; h < H_; ++h)
    atomicAdd(&ws[OFF_Z + ((b * H_ + h) << 8) + d], z[h]);
}

// ---- K5: t = m2 + log2(Z) ----------------------------------------------------
__global__ void k5_t(float* __restrict__ ws) {
  int i = blockIdx.x * blockDim.x + threadIdx.x;
  if (i >= B_ * H_ * D_) return;
  ws[OFF_T + i] = ws[OFF_M2 + i] + fast_log2(ws[OFF_Z + i]);
}

// ---- K6: out[b,s,d] = max(Vsum*Amax, Vsum*Amin), A_h = Ac[h]*Qs[b,h,s] ------
#define K6_ROWS 32
__global__ void k6_out(const float* __restrict__ q, float* __restrict__ out,
                       const float* __restrict__ ws) {
  const int rowBlocks = S_ / K6_ROWS;
  const int b  = blockIdx.x / rowBlocks;
  const int s0 = (blockIdx.x % rowBlocks) * K6_ROWS;
  const int d  = threadIdx.x;
  const int wave = threadIdx.x >> 5, lane = threadIdx.x & 31;
  float K[H_], tl[H_], Ac[H_];
#pragma unroll
  for (int h = 0; h < H_; ++h) {
    K[h]  = ws[OFF_K + h];
    tl[h] = ws[OFF_T + ((b * H_ + h) << 8) + d];
    Ac[h] = ws[OFF_AC + h];
  }
  const float Vs = ws[OFF_VSUM + (b << 8) + d];
  __shared__ float wsum[2][8][H_];
  const float* qp = q + ((size_t)(b * S_ + s0)) * D_ + d;
  float* op = out + ((size_t)(b * S_ + s0)) * D_ + d;
  for (int r = 0; r < K6_ROWS; ++r) {
    if (r + 4 < K6_ROWS) __builtin_prefetch(qp + (size_t)(r + 4) * D_, 0, 1);
    float qv = qp[(size_t)r * D_];
    float eh[H_];
#pragma unroll
    for (int h = 0; h < H_; ++h) eh[h] = fast_exp2(fmaf(K[h], qv, -tl[h]));
    // full wave butterfly: all lanes end with the wave-sum for each h
#pragma unroll
    for (int h = 0; h < H_; ++h) {
#pragma unroll
      for (int o = 16; o; o >>= 1) eh[h] += __shfl_xor(eh[h], o, 32);
    }
    const int p = r & 1;
    if (lane == 0) {
#pragma unroll
      for (int h = 0; h < H_; ++h) wsum[p][wave][h] = eh[h];
    }
    __syncthreads();
    // every wave combines the 8 wave-partials (double-buffered => 1 barrier/row)
    const int hh = lane & 7, w0 = lane >> 3;  // w0 in 0..3
    float v = wsum[p][w0][hh] + wsum[p][w0 + 4][hh];
    v += __shfl_xor(v, 8, 32);
    v += __shfl_xor(v, 16, 32);               // v = Qs[b,hh,s]
    float A = Ac[hh] * v;
    float amx = A, amn = A;
#pragma unroll
    for (int o = 1; o < 8; o <<= 1) {
      amx = fmaxf(amx, __shfl_xor(amx, o, 32));
      amn = fminf(amn, __shfl_xor(amn, o, 32));
    }
    op[(size_t)r * D_] = fmaxf(Vs * amx, Vs * amn);
  }
}

extern "C" void kernel_launch(void* const* d_in, const int* in_sizes, int n_in,
                              void* d_out, int out_size, void* d_ws, size_t ws_size,
                              hipStream_t stream) {
  const float* q  = (const float*)d_in[0];
  // d_in[1] (k) and d_in[4] (k_weights) are provably unused by the reference.
  const float* v  = (const float*)d_in[2];
  const float* qw = (const float*)d_in[3];
  const float* vw = (const float*)d_in[5];
  float* out = (float*)d_out;
  float* ws  = (float*)d_ws;

  k0_init<<<(OFF_T + 255) / 256, 256, 0, stream>>>(ws);
  k1_wsum<<<16, 256, 0, stream>>>(qw, vw, ws);
  k2_stats<<<B_ * K2_CHUNKS, 256, 0, stream>>>(q, v, ws);
  k3_prep<<<(B_ * H_ * D_ + 255) / 256, 256, 0, stream>>>(ws);
  k4_z<<<B_ * K4_CHUNKS, 256, 0, stream>>>(q, ws);
  k5_t<<<(B_ * H_ * D_ + 255) / 256, 256, 0, stream>>>(ws);
  k6_out<<<B_ * (S_ / K6_ROWS), 256, 0, stream>>>(q, out, ws);
}